// SecondaryInputLayer_57921928953979
// MI455X (gfx1250) — compile-verified
//
#include <hip/hip_runtime.h>

// out[b, j] = x[b, j] * w[j, j]   (x: 8192x4096 f32, w: 4096x4096 f32)
// Memory-bound: ~268 MB of traffic -> ~11.5 us at 23.3 TB/s.
// Strategy: per-block async-DMA the needed diagonal slice into LDS (CDNA5
// global_load_async_to_lds + s_wait_asynccnt), hoist each thread's 4 diag
// values into registers, then stream x -> out with 128-bit non-temporal
// loads/stores.

#define UNITS 4096
#define BATCH 8192
#define COLS_PER_BLOCK 1024   // columns handled by one 256-thread block
#define TPB 256               // 8 waves (wave32) per block
#define ROW_BLOCKS 2048       // grid.y; each block streams BATCH/ROW_BLOCKS = 4 rows

typedef float v4f __attribute__((ext_vector_type(4)));

typedef __attribute__((address_space(1))) int* gptr_i32;
typedef __attribute__((address_space(3))) int* lptr_i32;

#ifdef __AMDGCN__
#if __has_builtin(__builtin_amdgcn_global_load_async_to_lds_b32)
#define HAVE_ASYNC_LDS 1
#endif
#endif
#ifndef HAVE_ASYNC_LDS
#define HAVE_ASYNC_LDS 0
#endif

__global__ __launch_bounds__(TPB) void diag_scale_kernel(
    const float* __restrict__ x, const float* __restrict__ w,
    float* __restrict__ out) {
  __shared__ __align__(16) float sdiag[COLS_PER_BLOCK];

  const int t = threadIdx.x;
  const int colBase = (int)blockIdx.x * COLS_PER_BLOCK;
  const int col = colBase + t * 4;  // this thread's 4 columns (fixed)

  // ---- Stage diag(w) slice for this block's columns into LDS ----
  // diag[c] = w[c * (UNITS+1)]  (strided gather; 16 KiB total, L2-resident)
#if HAVE_ASYNC_LDS
#pragma unroll
  for (int k = 0; k < 4; ++k) {
    const float* gp = w + (size_t)(col + k) * (size_t)(UNITS + 1);
    __builtin_amdgcn_global_load_async_to_lds_b32(
        (gptr_i32)(size_t)gp,
        (lptr_i32)&sdiag[t * 4 + k],
        /*offset=*/0, /*cpol=*/0);
  }
#if __has_builtin(__builtin_amdgcn_s_wait_asynccnt)
  __builtin_amdgcn_s_wait_asynccnt(0);
#else
  asm volatile("s_wait_asynccnt 0" ::: "memory");
#endif
#else
#pragma unroll
  for (int k = 0; k < 4; ++k) {
    sdiag[t * 4 + k] = w[(size_t)(col + k) * (size_t)(UNITS + 1)];
  }
#endif
  __syncthreads();

  // Hoist the loop-invariant scale into registers.
  const v4f d4 = *(const v4f*)&sdiag[t * 4];

  // ---- Stream rows: 128-bit coalesced, non-temporal (single-use data) ----
  for (int row = (int)blockIdx.y; row < BATCH; row += ROW_BLOCKS) {
    const size_t off = (size_t)row * UNITS + (size_t)col;
    v4f v = __builtin_nontemporal_load((const v4f*)(x + off));
    v *= d4;
    __builtin_nontemporal_store(v, (v4f*)(out + off));
  }
}

extern "C" void kernel_launch(void* const* d_in, const int* in_sizes, int n_in,
                              void* d_out, int out_size, void* d_ws,
                              size_t ws_size, hipStream_t stream) {
  (void)in_sizes; (void)n_in; (void)out_size; (void)d_ws; (void)ws_size;
  const float* x = (const float*)d_in[0];
  const float* w = (const float*)d_in[1];
  float* out = (float*)d_out;

  dim3 grid(UNITS / COLS_PER_BLOCK, ROW_BLOCKS, 1);  // 4 x 2048 = 8192 blocks
  diag_scale_kernel<<<grid, TPB, 0, stream>>>(x, w, out);
}